// GraphAttentionLayer_541165879455
// MI455X (gfx1250) — compile-verified
//
#include <hip/hip_runtime.h>

// Problem constants (match reference)
#define Nn 100000
#define Ee 1600000
#define Ff 128
#define Hh 4
#define HDd 32
#define EPSc 1e-3f

typedef float v2f __attribute__((ext_vector_type(2)));
typedef float v8f __attribute__((ext_vector_type(8)));

__device__ __forceinline__ v8f wmma_f32(v2f a, v2f b, v8f c) {
  // V_WMMA_F32_16X16X4_F32 : D = A(16x4) * B(4x16) + C(16x16), fp32 exact
  return __builtin_amdgcn_wmma_f32_16x16x4_f32(false, a, false, b, (short)0, c, false, false);
}

// ---- CDNA5 async global->LDS copy (ASYNCcnt-tracked) ----------------------
// LDS byte offset = low 32 bits of the flat shared-aperture address
// (ISA 10.2: LDS aperture maps addr[31:0] to the LDS offset).
__device__ __forceinline__ unsigned lds_off(const void* p) {
  return (unsigned)(unsigned long long)p;
}
__device__ __forceinline__ void async_load_b128(unsigned lds_byte, const void* gaddr) {
  asm volatile("global_load_async_to_lds_b128 %0, %1, off"
               :: "v"(lds_byte), "v"(gaddr) : "memory");
}
template <int N>
__device__ __forceinline__ void wait_asynccnt() {
#if __has_builtin(__builtin_amdgcn_s_wait_asynccnt)
  __builtin_amdgcn_s_wait_asynccnt(N);
  asm volatile("" ::: "memory");
#else
  asm volatile("s_wait_asynccnt %0" :: "n"(N) : "memory");
#endif
}

// monotonic float<->uint encoding for atomicMax-based float max
__device__ __forceinline__ unsigned encf(float f) {
  unsigned u = __float_as_uint(f);
  return (u & 0x80000000u) ? ~u : (u | 0x80000000u);
}
__device__ __forceinline__ float decf(unsigned k) {
  return (k & 0x80000000u) ? __uint_as_float(k & 0x7FFFFFFFu) : __uint_as_float(~k);
}

// ---------------------------------------------------------------------------
// K0: init per-head max/sum slots
// ---------------------------------------------------------------------------
__global__ void init_stats_kernel(unsigned* __restrict__ hmax, float* __restrict__ hsum) {
  int t = threadIdx.x;
  if (t < Hh) { hmax[t] = 0u; hsum[t] = 0.0f; }
}

// ---------------------------------------------------------------------------
// K1: q/k/v = x @ W  (blockIdx.y selects which W / destination)
// 8 waves; wave w owns output column tile [16w,16w+16). B fragments
// (32 x v2f = 64 VGPRs) preloaded once. A tile double-buffered in LDS,
// filled by async global->LDS copies overlapped with the WMMA chain.
// ---------------------------------------------------------------------------
__global__ __launch_bounds__(256) void qkv_gemm_kernel(
    const float* __restrict__ x,
    const float* __restrict__ Wq, const float* __restrict__ Wk, const float* __restrict__ Wv,
    float* __restrict__ qo, float* __restrict__ ko, float* __restrict__ vo,
    int rowTiles) {
  const float* W; float* dst;
  if (blockIdx.y == 0)      { W = Wq; dst = qo; }
  else if (blockIdx.y == 1) { W = Wk; dst = ko; }
  else                      { W = Wv; dst = vo; }

  __shared__ float xs[2][16 * 132];

  const int tid  = threadIdx.x;
  const int wave = tid >> 5;
  const int lane = tid & 31;
  const int hi   = lane >> 4;     // half-wave selects K pair
  const int ln   = lane & 15;     // row (A) / col (B,D) within tile
  const int col  = wave * 16 + ln;

  // Preload all 32 B fragments for this wave's column tile (W is L2-hot).
  v2f bf[32];
#pragma unroll
  for (int kk = 0; kk < 32; ++kk) {
    const int k0 = 4 * kk + 2 * hi;
    bf[kk].x = W[k0 * Ff + col];
    bf[kk].y = W[(k0 + 1) * Ff + col];
  }

  // per-thread copy geometry: 2x b128 per tile
  const int cr  = tid >> 5;          // rows 0..7 (and +8)
  const int cc4 = (tid & 31) * 4;    // col group
  const unsigned l0[2] = { lds_off(&xs[0][cr * 132 + cc4]),
                           lds_off(&xs[1][cr * 132 + cc4]) };
  const unsigned l1[2] = { lds_off(&xs[0][(cr + 8) * 132 + cc4]),
                           lds_off(&xs[1][(cr + 8) * 132 + cc4]) };

  auto issue_tile = [&](int tile, int buf) {
    const float* g0 = &x[(size_t)(tile * 16 + cr) * Ff + cc4];
    const float* g1 = &x[(size_t)(tile * 16 + cr + 8) * Ff + cc4];
    async_load_b128(l0[buf], g0);
    async_load_b128(l1[buf], g1);
  };

  int t = blockIdx.x;
  int buf = 0;
  if (t < rowTiles) issue_tile(t, 0);

  for (; t < rowTiles; t += gridDim.x) {
    const int tn = t + gridDim.x;
    if (tn < rowTiles) { issue_tile(tn, buf ^ 1); wait_asynccnt<2>(); }
    else               { wait_asynccnt<0>(); }
    __syncthreads();

    const float* xb = xs[buf];
    v8f acc = {};
#pragma unroll
    for (int kk = 0; kk < 32; ++kk) {
      v2f a = *(const v2f*)&xb[ln * 132 + kk * 4 + 2 * hi];
      acc = wmma_f32(a, bf[kk], acc);
    }
    const int row0 = t * 16;
#pragma unroll
    for (int i = 0; i < 8; ++i) {
      const int row = row0 + i + 8 * hi;   // D layout: VGPR i -> M=i (+8 for hi half)
      dst[(size_t)row * Ff + col] = acc[i];
    }
    __syncthreads();      // all reads of xs[buf] done before it is refilled
    buf ^= 1;
  }
}

// ---------------------------------------------------------------------------
// K2: scores[e,h] = dot(q[dst,h,:], k[src,h,:] + emb[type,:]) / sqrt(32)
//     + per-head global max via encoded atomicMax
// ---------------------------------------------------------------------------
__global__ __launch_bounds__(256) void scores_kernel(
    const float* __restrict__ q, const float* __restrict__ k,
    const int* __restrict__ ei, const int* __restrict__ et,
    const float* __restrict__ emb,
    float* __restrict__ p, unsigned* __restrict__ hmax) {
  __shared__ unsigned smax[Hh];
  const int tid = threadIdx.x;
  if (tid < Hh) smax[tid] = 0u;
  __syncthreads();

  const unsigned g = blockIdx.x * 256u + tid;   // g in [0, E*H)
  const int e = (int)(g >> 2);
  const int h = (int)(g & 3);
  const int src = ei[e];
  const int dst = ei[Ee + e];
  const int ty  = et[e];

  const float4* qp = (const float4*)(q + (size_t)dst * Ff + h * HDd);
  const float4* kp = (const float4*)(k + (size_t)src * Ff + h * HDd);
  const float4* ep = (const float4*)(emb + ty * HDd);
  float s = 0.0f;
#pragma unroll
  for (int d = 0; d < 8; ++d) {
    float4 qv = qp[d], kv = kp[d], ev = ep[d];
    s += qv.x * (kv.x + ev.x) + qv.y * (kv.y + ev.y) +
         qv.z * (kv.z + ev.z) + qv.w * (kv.w + ev.w);
  }
  s *= 0.17677669529663687f;   // 1/sqrt(32)
  p[g] = s;

  atomicMax(&smax[h], encf(s));
  __syncthreads();
  if (tid < Hh) atomicMax(&hmax[tid], smax[tid]);
}

// ---------------------------------------------------------------------------
// K3: p = exp(score - max[h]); per-head sum via atomicAdd
// ---------------------------------------------------------------------------
__global__ __launch_bounds__(256) void expsum_kernel(
    float* __restrict__ p, const unsigned* __restrict__ hmax,
    float* __restrict__ hsum) {
  __shared__ float ssum[Hh];
  const int tid = threadIdx.x;
  if (tid < Hh) ssum[tid] = 0.0f;
  __syncthreads();

  const unsigned g = blockIdx.x * 256u + tid;
  const int h = (int)(g & 3);
  const float mx = decf(hmax[h]);
  const float pe = __expf(p[g] - mx);
  p[g] = pe;

  atomicAdd(&ssum[h], pe);
  __syncthreads();
  if (tid < Hh) atomicAdd(&hsum[tid], ssum[tid]);
}

// ---------------------------------------------------------------------------
// K4: zero the aggregation buffer (aliased onto q workspace)
// ---------------------------------------------------------------------------
__global__ __launch_bounds__(256) void zero_kernel(float4* __restrict__ buf) {
  const unsigned g = blockIdx.x * 256u + threadIdx.x;
  buf[g] = make_float4(0.f, 0.f, 0.f, 0.f);
}

// ---------------------------------------------------------------------------
// K5: agg[dst, c] += (p[e,h]/sum[h]) * v[src, c]
// one thread per (edge, 4 channels): b128 gather of v, 4x atomic f32 add.
// agg (51 MB) and v (51 MB) are L2-resident (192 MB L2).
// ---------------------------------------------------------------------------
__global__ __launch_bounds__(256) void aggregate_kernel(
    const float* __restrict__ p, const float* __restrict__ hsum,
    const float* __restrict__ v, const int* __restrict__ ei,
    float* __restrict__ agg) {
  const unsigned g = blockIdx.x * 256u + threadIdx.x;  // [0, E*32)
  const int e  = (int)(g >> 5);
  const int c4 = (int)(g & 31) * 4;       // channel base
  const int h  = c4 >> 5;
  const int src = ei[e];
  const int dst = ei[Ee + e];
  const float attn = p[(e << 2) + h] / hsum[h];
  const float4 vv = *(const float4*)&v[(size_t)src * Ff + c4];
  float* ap = &agg[(size_t)dst * Ff + c4];
  atomicAdd(ap + 0, attn * vv.x);
  atomicAdd(ap + 1, attn * vv.y);
  atomicAdd(ap + 2, attn * vv.z);
  atomicAdd(ap + 3, attn * vv.w);
}

// ---------------------------------------------------------------------------
// K6: out = LayerNorm(agg @ Wo + bo + x) * gamma + beta
// Double-buffered async-LDS WMMA GEMM, fused bias/residual/LayerNorm epilogue.
// ---------------------------------------------------------------------------
__global__ __launch_bounds__(256) void out_kernel(
    const float* __restrict__ agg, const float* __restrict__ Wo,
    const float* __restrict__ bo, const float* __restrict__ x,
    const float* __restrict__ gamma, const float* __restrict__ beta,
    float* __restrict__ out, int rowTiles) {
  __shared__ float xs[2][16 * 132];
  __shared__ float ys[16 * 132];
  __shared__ float ps[16][17];
  __shared__ float pq[16][17];
  __shared__ float mv[16][2];

  const int tid  = threadIdx.x;
  const int wave = tid >> 5;
  const int lane = tid & 31;
  const int hi   = lane >> 4;
  const int ln   = lane & 15;
  const int col  = wave * 16 + ln;

  v2f bf[32];
#pragma unroll
  for (int kk = 0; kk < 32; ++kk) {
    const int k0 = 4 * kk + 2 * hi;
    bf[kk].x = Wo[k0 * Ff + col];
    bf[kk].y = Wo[(k0 + 1) * Ff + col];
  }
  const float bias = bo[col];

  const int cr  = tid >> 5;
  const int cc4 = (tid & 31) * 4;
  const unsigned l0[2] = { lds_off(&xs[0][cr * 132 + cc4]),
                           lds_off(&xs[1][cr * 132 + cc4]) };
  const unsigned l1[2] = { lds_off(&xs[0][(cr + 8) * 132 + cc4]),
                           lds_off(&xs[1][(cr + 8) * 132 + cc4]) };

  auto issue_tile = [&](int tile, int buf) {
    const float* g0 = &agg[(size_t)(tile * 16 + cr) * Ff + cc4];
    const float* g1 = &agg[(size_t)(tile * 16 + cr + 8) * Ff + cc4];
    async_load_b128(l0[buf], g0);
    async_load_b128(l1[buf], g1);
  };

  int t = blockIdx.x;
  int buf = 0;
  if (t < rowTiles) issue_tile(t, 0);

  for (; t < rowTiles; t += gridDim.x) {
    const int tn = t + gridDim.x;
    if (tn < rowTiles) { issue_tile(tn, buf ^ 1); wait_asynccnt<2>(); }
    else               { wait_asynccnt<0>(); }
    __syncthreads();

    const float* xb = xs[buf];
    v8f acc = {};
#pragma unroll
    for (int kk = 0; kk < 32; ++kk) {
      v2f a = *(const v2f*)&xb[ln * 132 + kk * 4 + 2 * hi];
      acc = wmma_f32(a, bf[kk], acc);
    }
    const int row0 = t * 16;
    // bias + residual into LDS row-tile
#pragma unroll
    for (int i = 0; i < 8; ++i) {
      const int lr = i + 8 * hi;
      ys[lr * 132 + col] = acc[i] + bias + x[(size_t)(row0 + lr) * Ff + col];
    }
    __syncthreads();

    // LayerNorm: 16 rows x 16 groups of 8 channels
    const int row = tid >> 4;
    const int grp = tid & 15;
    {
      float s = 0.f, sq = 0.f;
#pragma unroll
      for (int j = 0; j < 8; ++j) {
        float yv = ys[row * 132 + grp * 8 + j];
        s += yv; sq += yv * yv;
      }
      ps[row][grp] = s; pq[row][grp] = sq;
    }
    __syncthreads();
    if (grp == 0) {
      float s = 0.f, sq = 0.f;
#pragma unroll
      for (int j = 0; j < 16; ++j) { s += ps[row][j]; sq += pq[row][j]; }
      const float mu  = s * (1.0f / Ff);
      const float var = sq * (1.0f / Ff) - mu * mu;
      mv[row][0] = mu;
      mv[row][1] = rsqrtf(var + EPSc);
    }
    __syncthreads();
    {
      const float mu = mv[row][0];
      const float rs = mv[row][1];
#pragma unroll
      for (int j = 0; j < 8; ++j) {
        const int c = grp * 8 + j;
        out[(size_t)(row0 + row) * Ff + c] =
            gamma[c] * (ys[row * 132 + c] - mu) * rs + beta[c];
      }
    }
    __syncthreads();      // ys/mv consumed; xs[buf] free for refill
    buf ^= 1;
  }
}

// ---------------------------------------------------------------------------
extern "C" void kernel_launch(void* const* d_in, const int* in_sizes, int n_in,
                              void* d_out, int out_size, void* d_ws, size_t ws_size,
                              hipStream_t stream) {
  const float* x     = (const float*)d_in[0];
  const int*   ei    = (const int*)  d_in[1];
  const int*   et    = (const int*)  d_in[2];
  const float* Wq    = (const float*)d_in[3];
  const float* Wk    = (const float*)d_in[4];
  const float* Wv    = (const float*)d_in[5];
  const float* emb   = (const float*)d_in[6];
  const float* Wo    = (const float*)d_in[7];
  const float* bo    = (const float*)d_in[8];
  const float* gamma = (const float*)d_in[9];
  const float* beta  = (const float*)d_in[10];
  float* out = (float*)d_out;

  // workspace layout
  const size_t MAT = (size_t)Nn * Ff * sizeof(float);        // 51.2 MB
  char* ws = (char*)d_ws;
  float* qbuf = (float*)(ws);             // also reused as agg after scores
  float* kbuf = (float*)(ws + MAT);
  float* vbuf = (float*)(ws + 2 * MAT);
  float* pbuf = (float*)(ws + 3 * MAT);   // E*H probs, 25.6 MB
  unsigned* hmax = (unsigned*)(ws + 3 * MAT + (size_t)Ee * Hh * sizeof(float));
  float*    hsum = (float*)(hmax + 8);

  const int rowTiles = Nn / 16;           // 6250, exact

  init_stats_kernel<<<1, 32, 0, stream>>>(hmax, hsum);

  qkv_gemm_kernel<<<dim3(640, 3), 256, 0, stream>>>(x, Wq, Wk, Wv,
                                                    qbuf, kbuf, vbuf, rowTiles);

  scores_kernel<<<(Ee * Hh) / 256, 256, 0, stream>>>(qbuf, kbuf, ei, et, emb,
                                                     pbuf, hmax);

  expsum_kernel<<<(Ee * Hh) / 256, 256, 0, stream>>>(pbuf, hmax, hsum);

  // zero agg (aliases qbuf; q is dead after scores_kernel)
  zero_kernel<<<(Nn * Ff / 4) / 256, 256, 0, stream>>>((float4*)qbuf);

  aggregate_kernel<<<(Ee * 32) / 256, 256, 0, stream>>>(pbuf, hsum, vbuf, ei, qbuf);

  out_kernel<<<640, 256, 0, stream>>>(qbuf, Wo, bo, x, gamma, beta, out, rowTiles);
}